// GINLayer_79362405696145
// MI455X (gfx1250) — compile-verified
//
#include <hip/hip_runtime.h>

typedef __attribute__((ext_vector_type(2))) float v2f;
typedef __attribute__((ext_vector_type(8))) float v8f;

#define N_NODES 100000
#define N_EDGES 1600000
#define D 128

// ---------------------------------------------------------------------------
// Native CDNA5 device-scope f32 atomic add (no return -> STOREcnt tracked).
// Avoids any chance of a global_atomic_cmpswap CAS-loop expansion; the
// scatter phase is ~205M of these, so the native path matters most here.
// ---------------------------------------------------------------------------
__device__ __forceinline__ void atomic_add_f32_dev(float* p, float v) {
    asm volatile("global_atomic_add_f32 %0, %1, off scope:SCOPE_DEV"
                 :: "v"(p), "v"(v)
                 : "memory");
}

// ---------------------------------------------------------------------------
// Zero-init X_agg (float4 stores; d_ws is poisoned to 0xAA by the harness).
// ---------------------------------------------------------------------------
__global__ void GIN_zero_kernel(float4* __restrict__ p, int n4) {
    int i = blockIdx.x * blockDim.x + threadIdx.x;
    if (i < n4) p[i] = float4{0.0f, 0.0f, 0.0f, 0.0f};
}

// ---------------------------------------------------------------------------
// Edge scatter: X_agg[ref_b[e]] += X[ref_a[e]].
// One wave (32 lanes) per edge; each lane moves a float4 (512B/wave gather)
// and issues 4 native global_atomic_add_f32. X (51.2 MB) and X_agg are
// L2-resident (192 MB L2), so this is an L2-bandwidth/atomic-rate problem.
// ---------------------------------------------------------------------------
__global__ void __launch_bounds__(256)
GIN_scatter_kernel(const float* __restrict__ X,
                   const int* __restrict__ ref_a,
                   const int* __restrict__ ref_b,
                   float* __restrict__ Xagg) {
    long long tid = (long long)blockIdx.x * blockDim.x + threadIdx.x;
    int edge = (int)(tid >> 5);
    int lane = (int)(tid & 31);
    if (edge >= N_EDGES) return;
    int src = ref_a[edge];
    int dst = ref_b[edge];
    const float4 v = *(const float4*)(X + (size_t)src * D + lane * 4);
    float* o = Xagg + (size_t)dst * D + lane * 4;
    atomic_add_f32_dev(o + 0, v.x);
    atomic_add_f32_dev(o + 1, v.y);
    atomic_add_f32_dev(o + 2, v.z);
    atomic_add_f32_dev(o + 3, v.w);
}

// ---------------------------------------------------------------------------
// FP32 WMMA GEMM: out[M x 128] = (RELU?)(A[M x 128] @ W[128 x 128] + bias).
// One wave per 16-row strip: 8 accumulators (16x16 tiles across N=128),
// K-loop of 32 steps; each step loads ONE A fragment (float2/lane) and
// feeds 8 x V_WMMA_F32_16X16X4_F32 -> 8x A reuse, A traffic at its minimum.
//
// Fragment layouts (CDNA5 ISA, 32-bit operands, wave32):
//   A 16x4 : lanes 0-15 -> M=lane, VGPR0=K0, VGPR1=K1; lanes 16-31 -> K2,K3
//   B 4x16 : lanes 0-15 -> N=lane, VGPR0=K0, VGPR1=K1; lanes 16-31 -> K2,K3
//   C/D    : VGPR r, lanes 0-15 -> M=r, N=lane; lanes 16-31 -> M=r+8
// ---------------------------------------------------------------------------
template <bool RELU>
__global__ void __launch_bounds__(256)
GIN_gemm_wmma_kernel(const float* __restrict__ A,
                     const float* __restrict__ W,
                     const float* __restrict__ bias,
                     float* __restrict__ out) {
    const int wave  = threadIdx.x >> 5;
    const int lane  = threadIdx.x & 31;
    const int strip = blockIdx.x * (blockDim.x >> 5) + wave;  // 16-row strip
    const int row0  = strip * 16;
    if (row0 >= N_NODES) return;

    const int half = lane >> 4;                // 0: K pair {0,1}, 1: {2,3}
    const int m    = lane & 15;                // row (A) / col (B,C,D) index

    const float* arow  = A + (size_t)(row0 + m) * D + half * 2;  // A[row][k+2h]
    const float* wbase = W + (size_t)(half * 2) * D + m;         // W[k+2h][m]

    v8f acc[8];
    #pragma unroll
    for (int n = 0; n < 8; ++n) acc[n] = (v8f){0,0,0,0,0,0,0,0};

    for (int k = 0; k < D; k += 4) {
        const float2 av = *(const float2*)(arow + k);            // 8B-aligned
        v2f a; a.x = av.x; a.y = av.y;
        const float* wp = wbase + (size_t)k * D;
        #pragma unroll
        for (int n = 0; n < 8; ++n) {
            v2f b; b.x = wp[n * 16]; b.y = wp[n * 16 + D];
            acc[n] = __builtin_amdgcn_wmma_f32_16x16x4_f32(
                /*neg_a=*/false, a, /*neg_b=*/false, b,
                /*c_mod=*/(short)0, acc[n], /*reuse_a=*/false, /*reuse_b=*/false);
        }
    }

    #pragma unroll
    for (int n = 0; n < 8; ++n) {
        const float bv = bias[n * 16 + m];
        float* orow = out + (size_t)(row0 + half * 8) * D + n * 16 + m;
        #pragma unroll
        for (int r = 0; r < 8; ++r) {
            float v = acc[n][r] + bv;
            if (RELU) v = fmaxf(v, 0.0f);
            orow[(size_t)r * D] = v;
        }
    }
}

// ---------------------------------------------------------------------------
// Launch
// ---------------------------------------------------------------------------
extern "C" void kernel_launch(void* const* d_in, const int* in_sizes, int n_in,
                              void* d_out, int out_size, void* d_ws, size_t ws_size,
                              hipStream_t stream) {
    const float* X     = (const float*)d_in[0];
    const int*   ref_a = (const int*)d_in[1];
    const int*   ref_b = (const int*)d_in[2];
    const float* W_h   = (const float*)d_in[3];
    const float* b_h   = (const float*)d_in[4];
    const float* W_o   = (const float*)d_in[5];
    const float* b_o   = (const float*)d_in[6];
    float* out = (float*)d_out;

    float* Xagg = (float*)d_ws;                         // 51.2 MB
    float* Xhid = Xagg + (size_t)N_NODES * D;           // 51.2 MB

    // 1) zero X_agg
    const int n4 = N_NODES * D / 4;                     // 3.2M float4
    GIN_zero_kernel<<<(n4 + 255) / 256, 256, 0, stream>>>((float4*)Xagg, n4);

    // 2) edge scatter-add (one wave per edge, native f32 atomics)
    const long long sthreads = (long long)N_EDGES * 32;
    GIN_scatter_kernel<<<(unsigned)((sthreads + 255) / 256), 256, 0, stream>>>(
        X, ref_a, ref_b, Xagg);

    // 3) X_hid = X_agg @ W_h + b_h   (6250 strips, 8 waves/block)
    const int strips = N_NODES / 16;                    // 6250
    const int blocks = (strips + 7) / 8;                // 782
    GIN_gemm_wmma_kernel<false><<<blocks, 256, 0, stream>>>(Xagg, W_h, b_h, Xhid);

    // 4) out = relu(X_hid @ W_o + b_o)
    GIN_gemm_wmma_kernel<true><<<blocks, 256, 0, stream>>>(Xhid, W_o, b_o, out);
}